// Fusion_EdgeConv_43800076484862
// MI455X (gfx1250) — compile-verified
//
#include <hip/hip_runtime.h>
#include <float.h>

#define M_PTS 32768
#define D_FEAT 256
#define K_NBR 16

// Fragment-ready LDS strides (floats): pair-row length + 32-float pad so the
// two 16-lane halves of a wave land on disjoint LDS bank halves.
#define W1_PSTRIDE (2 * 128 + 32)   // 288 floats per row-pair, 256 pairs
#define W2_PSTRIDE (2 * 256 + 32)   // 544 floats per row-pair, 64 pairs

typedef float v2f __attribute__((ext_vector_type(2)));
typedef float v8f __attribute__((ext_vector_type(8)));

// Monotone order-preserving float<->uint mapping for atomicMax on floats.
__device__ __forceinline__ unsigned encf(float f) {
  unsigned u = __float_as_uint(f);
  return (u & 0x80000000u) ? ~u : (u | 0x80000000u);
}
__device__ __forceinline__ float decf(unsigned u) {
  return (u & 0x80000000u) ? __uint_as_float(u & 0x7FFFFFFFu)
                           : __uint_as_float(~u);
}

// ---------------------------------------------------------------- init ----
__global__ __launch_bounds__(256) void k_init(float* __restrict__ sumb,
                                              unsigned* __restrict__ maxb,
                                              int* __restrict__ cnt) {
  int gid = blockIdx.x * 256 + threadIdx.x;
  sumb[gid] = 0.0f;
  maxb[gid] = 0u;              // < encf of any finite float
  if (gid < M_PTS) cnt[gid] = 0;
}

// ------------------------------------------------- GEMM1: P|Q = F @ W1 ----
// PQ[i, 0:128]   = F[i] @ W1[0:256, :]          (src projection P)
// PQ[i, 128:256] = F[i] @ W1[256:512, :] + b1   (dst projection Q)
__global__ __launch_bounds__(256) void k_gemm1(const float* __restrict__ F,
                                               const float* __restrict__ W1,
                                               const float* __restrict__ b1,
                                               float* __restrict__ PQ) {
  extern __shared__ float lw[];  // W1 in fragment-ready pair layout (288 KB)
  for (int t = threadIdx.x; t < 512 * 128; t += 256) {
    int r = t >> 7, c = t & 127;
    lw[(r >> 1) * W1_PSTRIDE + 2 * c + (r & 1)] = W1[t];
  }
  __syncthreads();

  int wave = threadIdx.x >> 5, lane = threadIdx.x & 31;
  int half = lane >> 4, ln = lane & 15;
  int i0 = (blockIdx.x * 8 + wave) * 16;           // 16-row M tile
  int arow = i0 + ln;

  v8f acc[16];
  #pragma unroll
  for (int t = 0; t < 16; ++t) acc[t] = (v8f){};

  for (int s = 0; s < 64; ++s) {                   // K = 256, 4 per WMMA
    int kp = 2 * s + half;                         // row-pair (ko/2)
    float2 av = *(const float2*)(F + (size_t)arow * 256 + 4 * s + 2 * half);
    v2f afrag; afrag[0] = av.x; afrag[1] = av.y;
    const float* bp0 = lw + (size_t)kp * W1_PSTRIDE;          // P rows
    const float* bp1 = lw + (size_t)(128 + kp) * W1_PSTRIDE;  // Q rows
    #pragma unroll
    for (int t = 0; t < 16; ++t) {                 // N = 256, 16 tiles
      int nn = 16 * (t & 7) + ln;                  // column within half
      v2f bfrag = (t < 8) ? *(const v2f*)(bp0 + 2 * nn)
                          : *(const v2f*)(bp1 + 2 * nn);
      acc[t] = __builtin_amdgcn_wmma_f32_16x16x4_f32(
          false, afrag, false, bfrag, (short)0, acc[t], false, false);
    }
  }
  #pragma unroll
  for (int t = 0; t < 16; ++t) {
    int n = 16 * t + ln;
    #pragma unroll
    for (int r = 0; r < 8; ++r) {
      int m = r + 8 * half;
      float v = acc[t][r];
      if (t >= 8) v += b1[n - 128];                // bias only on Q half
      PQ[(size_t)(i0 + m) * 256 + n] = v;
    }
  }
}

// ------------------------------------------------------------- KNN -------
// One wave per query point: lane-strided scan with a per-lane sorted top-16,
// then a 16-round wave argmin merge through LDS.
__global__ __launch_bounds__(256) void k_knn(const float* __restrict__ coords,
                                             const int* __restrict__ bids,
                                             int* __restrict__ nbrs) {
  __shared__ float sd[8][512];
  __shared__ int si[8][512];
  int wave = threadIdx.x >> 5, lane = threadIdx.x & 31;
  int i = blockIdx.x * 8 + wave;

  float qx = coords[3 * i + 0], qy = coords[3 * i + 1], qz = coords[3 * i + 2];
  int qb = bids[i];

  float dl[16]; int il[16];
  #pragma unroll
  for (int t = 0; t < 16; ++t) { dl[t] = FLT_MAX; il[t] = 0; }

  for (int j = lane; j < M_PTS; j += 32) {
    float dx = coords[3 * j + 0] - qx;
    float dy = coords[3 * j + 1] - qy;
    float dz = coords[3 * j + 2] - qz;
    float d2 = dx * dx + dy * dy + dz * dz;
    bool valid = (bids[j] == qb) && (j != i);
    if (valid && d2 < dl[15]) {
      dl[15] = d2; il[15] = j;                      // one bubble pass
      #pragma unroll
      for (int t = 15; t >= 1; --t) {
        bool sw = dl[t] < dl[t - 1];
        float td = dl[t]; int ti = il[t];
        dl[t]     = sw ? dl[t - 1] : td;  il[t]     = sw ? il[t - 1] : ti;
        dl[t - 1] = sw ? td : dl[t - 1];  il[t - 1] = sw ? ti : il[t - 1];
      }
    }
  }
  #pragma unroll
  for (int t = 0; t < 16; ++t) {
    sd[wave][lane * 16 + t] = dl[t];
    si[wave][lane * 16 + t] = il[t];
  }
  __syncthreads();

  int p = 0;
  for (int r = 0; r < K_NBR; ++r) {
    float head = (p < 16) ? sd[wave][lane * 16 + p] : FLT_MAX;
    int hidx   = (p < 16) ? si[wave][lane * 16 + p] : 0;
    float mn = head;
    #pragma unroll
    for (int off = 16; off >= 1; off >>= 1)
      mn = fminf(mn, __shfl_xor(mn, off, 32));
    unsigned long long bal = __ballot(head == mn);
    int wl = (int)__ffsll(bal) - 1;
    int wi = __shfl(hidx, wl, 32);
    if (lane == wl) p++;
    if (lane == 0) nbrs[i * K_NBR + r] = wi;
  }
}

// ------------------------------- edge MLP + LayerNorm + scatter mean/max --
// One wave per center i: its 16 edges form one WMMA M-tile.
__global__ __launch_bounds__(256) void k_edge(const float* __restrict__ PQ,
                                              const int* __restrict__ nbrs,
                                              const float* __restrict__ W2,
                                              const float* __restrict__ b2,
                                              const float* __restrict__ lng,
                                              const float* __restrict__ lnb,
                                              float* __restrict__ sumb,
                                              unsigned* __restrict__ maxb,
                                              int* __restrict__ cnt) {
  extern __shared__ float lw[];  // W2 in fragment-ready pair layout (136 KB)
  for (int t = threadIdx.x; t < 128 * 256; t += 256) {
    int r = t >> 8, n = t & 255;
    lw[(r >> 1) * W2_PSTRIDE + 2 * n + (r & 1)] = W2[t];
  }
  __syncthreads();

  int wave = threadIdx.x >> 5, lane = threadIdx.x & 31;
  int half = lane >> 4, ln = lane & 15;
  int i = blockIdx.x * 8 + wave;

  int sv = nbrs[i * K_NBR + ln];                   // src of row (lane&15)
  const float* prow = PQ + (size_t)sv * 256;       // P[src], cols 0..127
  const float* qrow = PQ + (size_t)i * 256 + 128;  // Q[i]+b1, cols 0..127

  v8f acc[16];
  #pragma unroll
  for (int t = 0; t < 16; ++t) acc[t] = (v8f){};

  for (int s = 0; s < 32; ++s) {                   // K = 128
    int ko = 4 * s + 2 * half;
    float2 pv = *(const float2*)(prow + ko);
    float2 qv = *(const float2*)(qrow + ko);
    v2f afrag;
    afrag[0] = fmaxf(pv.x + qv.x, 0.0f);           // fused ReLU
    afrag[1] = fmaxf(pv.y + qv.y, 0.0f);
    const float* bp = lw + (size_t)(2 * s + half) * W2_PSTRIDE;
    #pragma unroll
    for (int t = 0; t < 16; ++t) {                 // N = 256
      v2f bfrag = *(const v2f*)(bp + 2 * (16 * t + ln));  // one ds_load_b64
      acc[t] = __builtin_amdgcn_wmma_f32_16x16x4_f32(
          false, afrag, false, bfrag, (short)0, acc[t], false, false);
    }
  }

  // + b2
  #pragma unroll
  for (int t = 0; t < 16; ++t) {
    float b2v = b2[16 * t + ln];
    #pragma unroll
    for (int r = 0; r < 8; ++r) acc[t][r] += b2v;
  }

  // LayerNorm stats per row (row m = r + 8*half, spread over 16 lanes)
  float rs[8], rq[8];
  #pragma unroll
  for (int r = 0; r < 8; ++r) { rs[r] = 0.0f; rq[r] = 0.0f; }
  #pragma unroll
  for (int t = 0; t < 16; ++t)
    #pragma unroll
    for (int r = 0; r < 8; ++r) {
      float v = acc[t][r]; rs[r] += v; rq[r] += v * v;
    }
  float mu[8], rstd[8];
  #pragma unroll
  for (int r = 0; r < 8; ++r) {
    float a = rs[r], b = rq[r];
    #pragma unroll
    for (int off = 8; off >= 1; off >>= 1) {       // stays within 16-lane half
      a += __shfl_xor(a, off, 32);
      b += __shfl_xor(b, off, 32);
    }
    mu[r] = a * (1.0f / 256.0f);
    float var = b * (1.0f / 256.0f) - mu[r] * mu[r];
    rstd[r] = rsqrtf(var + 1e-5f);
  }

  int srow[8];
  #pragma unroll
  for (int r = 0; r < 8; ++r) srow[r] = __shfl(sv, r + 8 * half, 32);

  // normalize + scatter (mean via atomicAdd, max via ordered-uint atomicMax)
  #pragma unroll
  for (int t = 0; t < 16; ++t) {
    int n = 16 * t + ln;
    float gv = lng[n], bb = lnb[n];
    #pragma unroll
    for (int r = 0; r < 8; ++r) {
      float h = (acc[t][r] - mu[r]) * rstd[r] * gv + bb;
      size_t addr = (size_t)srow[r] * 256 + n;
      atomicAdd(sumb + addr, h);
      atomicMax(maxb + addr, encf(h));
    }
  }
  if (lane < 16) atomicAdd(cnt + sv, 1);
}

// --------------------------------------------------------------- final ----
__global__ __launch_bounds__(256) void k_final(const float* __restrict__ sumb,
                                               const unsigned* __restrict__ maxb,
                                               const int* __restrict__ cnt,
                                               float* __restrict__ out) {
  int gid = blockIdx.x * 256 + threadIdx.x;
  int i = gid >> 8;
  int c = cnt[i];
  out[gid] = (c > 0) ? (sumb[gid] / (float)c + decf(maxb[gid])) : 0.0f;
}

// -------------------------------------------------------------------------
extern "C" void kernel_launch(void* const* d_in, const int* in_sizes, int n_in,
                              void* d_out, int out_size, void* d_ws,
                              size_t ws_size, hipStream_t stream) {
  (void)in_sizes; (void)n_in; (void)out_size; (void)ws_size;
  const float* F   = (const float*)d_in[0];
  const float* crd = (const float*)d_in[1];
  const int* bid   = (const int*)d_in[2];
  const float* W1  = (const float*)d_in[3];
  const float* b1  = (const float*)d_in[4];
  const float* W2  = (const float*)d_in[5];
  const float* b2  = (const float*)d_in[6];
  const float* lng = (const float*)d_in[7];
  const float* lnb = (const float*)d_in[8];
  float* out = (float*)d_out;

  // Workspace layout; d_out doubles as the running sum buffer.
  float* PQ      = (float*)d_ws;                          // M*256 f32
  unsigned* maxb = (unsigned*)(PQ + (size_t)M_PTS * 256); // M*256 u32
  int* cnt       = (int*)(maxb + (size_t)M_PTS * 256);    // M
  int* nbrs      = cnt + M_PTS;                           // M*K
  float* sumb    = out;                                   // M*256 f32

  const int lds_gemm1 = 256 * W1_PSTRIDE * 4;             // 294912 B
  const int lds_edge  = 64 * W2_PSTRIDE * 4;              // 139264 B
  hipFuncSetAttribute((const void*)k_gemm1,
                      hipFuncAttributeMaxDynamicSharedMemorySize, lds_gemm1);
  hipFuncSetAttribute((const void*)k_edge,
                      hipFuncAttributeMaxDynamicSharedMemorySize, lds_edge);

  k_init<<<M_PTS * 256 / 256, 256, 0, stream>>>(sumb, maxb, cnt);
  k_gemm1<<<M_PTS / (16 * 8), 256, lds_gemm1, stream>>>(F, W1, b1, PQ);
  k_knn<<<M_PTS / 8, 256, 0, stream>>>(crd, bid, nbrs);
  k_edge<<<M_PTS / 8, 256, lds_edge, stream>>>(PQ, nbrs, W2, b2, lng, lnb,
                                               sumb, maxb, cnt);
  k_final<<<M_PTS * 256 / 256, 256, 0, stream>>>(sumb, maxb, cnt, out);
}